// TopKActivation_51290499449025
// MI455X (gfx1250) — compile-verified
//
#include <hip/hip_runtime.h>
#include <stdint.h>

// Top-K (K=32) row mask-multiply for x[4096][32768] fp32.
// One 1024-thread workgroup per row:
//   A) async global->LDS stage of the 128KB row (CDNA5 ASYNCcnt path)
//   B) exact 32nd-largest value via MSB-first radix select; passes 2-3 run on a
//      compacted candidate list (top-byte bin survivors) instead of the full row
//   C) masked output, b128 LDS reads + nontemporal b128 global stores
// HBM traffic: one read + one write per element (~1.07 GB total, ~46us floor).

#define ROW_LEN   32768
#define NTHREADS  1024
#define TOPK      32
#define EPT       (ROW_LEN / NTHREADS)   // 32 elements per thread
#define CHUNKS    (EPT / 4)              // 8 x uint4 per thread
#define CAND_CAP  6144                   // candidate buffer (24 KB); uniform fallback if exceeded

typedef float  __attribute__((ext_vector_type(4))) floatx4;   // native vector for NT store

// Order-preserving key: larger key <=> larger float.
__device__ __forceinline__ uint32_t f2key(uint32_t u) {
    return (u & 0x80000000u) ? ~u : (u | 0x80000000u);
}

__global__ __launch_bounds__(NTHREADS)
void topk_mask_kernel(const float* __restrict__ x, float* __restrict__ out) {
    __shared__ __align__(16) uint32_t row[ROW_LEN];   // 128 KB raw fp32 bits
    __shared__ uint32_t cand[CAND_CAP];               // surviving keys after pass 1
    __shared__ uint32_t hist[260];                    // 256-bin histogram / suffix sums
    __shared__ uint32_t sel[4];                       // [0]=prefix [1]=remaining [2]=tie ctr [3]=cand count

    const int      tid  = threadIdx.x;
    const uint64_t base = (uint64_t)blockIdx.x * (uint64_t)ROW_LEN;
    const float*   grow = x + base;

    // ---------------- Phase A: async stage row into LDS --------------------------------
    #pragma unroll
    for (int c = 0; c < CHUNKS; ++c) {
        const int      elem   = (c * NTHREADS + tid) * 4;
        const uint32_t ldsOff = (uint32_t)(uintptr_t)(&row[elem]);   // low 32 bits = LDS byte addr
        const uint64_t ga     = (uint64_t)(uintptr_t)(grow + elem);
        asm volatile("global_load_async_to_lds_b128 %0, %1, off"
                     :: "v"(ldsOff), "v"(ga)
                     : "memory");
    }
#if __has_builtin(__builtin_amdgcn_s_wait_asynccnt)
    __builtin_amdgcn_s_wait_asynccnt(0);
#else
    asm volatile("s_wait_asynccnt 0" ::: "memory");
#endif
    __syncthreads();

    const uint4* row4 = reinterpret_cast<const uint4*>(row);
    uint32_t prefix    = 0;
    uint32_t remaining = TOPK;

    // ================= Pass 0: full-row histogram of key[31:24] ========================
    for (int i = tid; i < 260; i += NTHREADS) hist[i] = 0;
    __syncthreads();
    #pragma unroll
    for (int c = 0; c < CHUNKS; ++c) {
        const uint4 q = row4[c * NTHREADS + tid];
        atomicAdd(&hist[f2key(q.x) >> 24], 1u);
        atomicAdd(&hist[f2key(q.y) >> 24], 1u);
        atomicAdd(&hist[f2key(q.z) >> 24], 1u);
        atomicAdd(&hist[f2key(q.w) >> 24], 1u);
    }
    __syncthreads();
    // suffix scan + bin selection (shared by all passes)
    #define SELECT_BIN(SHIFT)                                                        \
        for (int step = 1; step < 256; step <<= 1) {                                 \
            uint32_t v = 0;                                                          \
            if (tid < 256) {                                                         \
                v = hist[tid];                                                       \
                if (tid + step < 256) v += hist[tid + step];                         \
            }                                                                        \
            __syncthreads();                                                         \
            if (tid < 256) hist[tid] = v;                                            \
            __syncthreads();                                                         \
        }                                                                            \
        if (tid < 256) {                                                             \
            const uint32_t suf     = hist[tid];                                      \
            const uint32_t sufNext = (tid == 255) ? 0u : hist[tid + 1];              \
            if (suf >= remaining && sufNext < remaining) {                           \
                sel[0] = prefix | ((uint32_t)tid << (SHIFT));                        \
                sel[1] = remaining - sufNext;                                        \
            }                                                                        \
        }                                                                            \
        __syncthreads();                                                             \
        prefix    = sel[0];                                                          \
        remaining = sel[1];                                                          \
        __syncthreads();
    SELECT_BIN(24)

    // ================= Pass 1: full-row scan, histogram key[23:16] + compact ===========
    for (int i = tid; i < 260; i += NTHREADS) hist[i] = 0;
    if (tid == 0) sel[3] = 0u;
    __syncthreads();
    {
        const uint32_t topByte = prefix >> 24;
        #pragma unroll
        for (int c = 0; c < CHUNKS; ++c) {
            const uint4 q = row4[c * NTHREADS + tid];
            uint32_t ks[4] = { f2key(q.x), f2key(q.y), f2key(q.z), f2key(q.w) };
            #pragma unroll
            for (int j = 0; j < 4; ++j) {
                const uint32_t key = ks[j];
                if ((key >> 24) == topByte) {
                    atomicAdd(&hist[(key >> 16) & 0xFFu], 1u);
                    const uint32_t pos = atomicAdd(&sel[3], 1u);
                    if (pos < CAND_CAP) cand[pos] = key;
                }
            }
        }
    }
    __syncthreads();
    SELECT_BIN(16)
    const uint32_t nCand = sel[3];      // stable after pass-1 scan barrier; uniform

    // ================= Passes 2,3: candidate-list scan (fallback: full row) ============
    for (int pass = 2; pass < 4; ++pass) {
        const int      shift  = 24 - 8 * pass;            // 8, then 0
        const uint32_t hiMask = 0xFFFFFFFFu << (shift + 8);
        for (int i = tid; i < 260; i += NTHREADS) hist[i] = 0;
        __syncthreads();
        if (nCand <= CAND_CAP) {                          // uniform branch
            for (uint32_t i = tid; i < nCand; i += NTHREADS) {
                const uint32_t key = cand[i];
                if ((key & hiMask) == (prefix & hiMask))
                    atomicAdd(&hist[(key >> shift) & 0xFFu], 1u);
            }
        } else {                                          // pathological row: full scan
            #pragma unroll
            for (int c = 0; c < CHUNKS; ++c) {
                const uint4 q = row4[c * NTHREADS + tid];
                uint32_t ks[4] = { f2key(q.x), f2key(q.y), f2key(q.z), f2key(q.w) };
                #pragma unroll
                for (int j = 0; j < 4; ++j) {
                    if ((ks[j] & hiMask) == (prefix & hiMask))
                        atomicAdd(&hist[(ks[j] >> shift) & 0xFFu], 1u);
                }
            }
        }
        __syncthreads();
        if (shift == 8) { SELECT_BIN(8) } else { SELECT_BIN(0) }
    }
    // prefix == exact key of K-th largest; 'remaining' == #equal keys to accept (>=1)

    if (tid == 0) sel[2] = 0u;          // tie-acceptance counter
    __syncthreads();

    // ---------------- Phase C: masked output, b128 in / nontemporal b128 out -----------
    float* orow = out + base;
    #pragma unroll
    for (int c = 0; c < CHUNKS; ++c) {
        const int   vidx = c * NTHREADS + tid;
        const uint4 q    = row4[vidx];
        uint32_t bs[4] = { q.x, q.y, q.z, q.w };
        floatx4  f4;
        #pragma unroll
        for (int j = 0; j < 4; ++j) {
            const uint32_t key = f2key(bs[j]);
            float o = 0.0f;
            if (key > prefix) {
                o = __uint_as_float(bs[j]);
            } else if (key == prefix) {
                if (atomicAdd(&sel[2], 1u) < remaining) o = __uint_as_float(bs[j]);
            }
            f4[j] = o;
        }
        __builtin_nontemporal_store(f4, reinterpret_cast<floatx4*>(orow) + vidx);
    }
}

extern "C" void kernel_launch(void* const* d_in, const int* in_sizes, int n_in,
                              void* d_out, int out_size, void* d_ws, size_t ws_size,
                              hipStream_t stream) {
    const float* x   = (const float*)d_in[0];
    float*       out = (float*)d_out;
    const int    rows = out_size / ROW_LEN;   // 4096
    topk_mask_kernel<<<rows, NTHREADS, 0, stream>>>(x, out);
}